// ColumnStep_71347996721416
// MI455X (gfx1250) — compile-verified
//
#include <hip/hip_runtime.h>
#include <math.h>
#include <stdint.h>

typedef __attribute__((ext_vector_type(2))) float v2f;
typedef __attribute__((ext_vector_type(8))) float v8f;

#define WMMA_F32(a,b,c) __builtin_amdgcn_wmma_f32_16x16x4_f32(false,(a),false,(b),(short)0,(c),false,false)

constexpr int Vv = 32000, Kk = 256, Bb = 2, Tt = 2048, NBr = 4, INNERd = 128;
constexpr int Mrows = Bb * Tt;            // 4096
constexpr float EPSf = 1.1920929e-07f;

// macro-tile config for the two big attention GEMMs
constexpr int BI = 128;                   // rows per block (8 waves x 16)
constexpr int BJ = 64;                    // score cols per block (4 subtiles per wave)
constexpr int BKc = 32;                   // K chunk staged in LDS
constexpr int LDA = BKc + 4;              // 36: row stride (floats), keeps B128 16B-aligned, spreads banks
constexpr int RBN = 64;                   // retrieved cols per block
constexpr int LDBR = RBN + 4;             // 68

__device__ __forceinline__ void async_b128(uint32_t lds_byte_off, uint64_t gaddr) {
  asm volatile("global_load_async_to_lds_b128 %0, %1, off" :: "v"(lds_byte_off), "v"(gaddr) : "memory");
}
__device__ __forceinline__ void wait_async() {
  asm volatile("s_wait_asynccnt 0x0" ::: "memory");
}
__device__ __forceinline__ uint32_t lds_off(const void* p) {
  // generic LDS pointer: low 32 bits are the offset within the wave's LDS allocation
  return (uint32_t)(uintptr_t)p;
}

// ---------------- gather + rmsnorm ----------------
__global__ void k_gather_norm(const float* __restrict__ x, const int* __restrict__ ridx,
                              float* __restrict__ g, float* __restrict__ gn) {
  int row = blockIdx.x;
  int k = threadIdx.x;                     // 0..255
  float val = x[(size_t)row * Vv + ridx[k]];
  __shared__ float red[256];
  red[k] = val * val;
  __syncthreads();
  for (int s = 128; s > 0; s >>= 1) { if (k < s) red[k] += red[k + s]; __syncthreads(); }
  float scale = rsqrtf(red[0] / (float)Kk + EPSf);
  g[(size_t)row * Kk + k] = val;
  gn[(size_t)row * Kk + k] = val * scale;
}

// ---------------- QKV projection: one wave -> one 16x16 tile ----------------
__global__ void k_qkv(const float* __restrict__ gn, const float* __restrict__ Wq,
                      const float* __restrict__ Wk, const float* __restrict__ Wv,
                      float* __restrict__ q, float* __restrict__ kk, float* __restrict__ vv) {
  const int Mt = Mrows / 16, Nt = Kk / 16;
  int id = blockIdx.x;
  int which = id / (Mt * Nt);
  int rem = id % (Mt * Nt);
  int mt = rem / Nt, nt = rem % Nt;
  const float* W = which == 0 ? Wq : (which == 1 ? Wk : Wv);
  float* out = which == 0 ? q : (which == 1 ? kk : vv);
  int l = threadIdx.x, half = l >> 4, lr = l & 15;
  int m = mt * 16 + lr, n = nt * 16 + lr;
  v8f acc = {};
  #pragma unroll 4
  for (int k0 = 0; k0 < Kk; k0 += 4) {
    int ka = k0 + 2 * half;
    v2f a;  a.x = gn[(size_t)m * Kk + ka]; a.y = gn[(size_t)m * Kk + ka + 1];
    v2f bf; bf.x = W[(size_t)ka * Kk + n]; bf.y = W[(size_t)(ka + 1) * Kk + n];
    acc = WMMA_F32(a, bf, acc);
  }
  #pragma unroll
  for (int r = 0; r < 8; r++) {
    int mm = mt * 16 + r + 8 * half;
    out[(size_t)mm * Kk + n] = acc[r];
  }
}

// ---------------- scores: SW = (Q K^T) * decay-weights / sqrt(K) ----------------
// LDS-tiled: 8 waves, macro 128(i) x 64(j), K chunked by 32, async global->LDS staging.
__global__ void __launch_bounds__(256) k_scores(const float* __restrict__ q,
                                                const float* __restrict__ kv,
                                                const float* __restrict__ dlogit,
                                                float* __restrict__ SW) {
  __shared__ float As[BI * LDA];           // 128 x 36 (q rows, k-contiguous)
  __shared__ float Bs[BJ * LDA];           // 64 x 36  (kk rows, k-contiguous)
  const int ITG = Tt / BI;                 // 16
  const int JTG = Tt / BJ;                 // 32
  int id = blockIdx.x;
  int b = id / (ITG * JTG);
  int rem = id % (ITG * JTG);
  int itg = rem / JTG, jtg = rem % JTG;
  int i0 = itg * BI, j0 = jtg * BJ;
  int t = threadIdx.x;
  size_t swbase = (size_t)b * Tt * Tt;

  if (j0 + BJ - 1 <= i0) {                 // whole macro-tile strictly at/below diagonal: zeros
    #pragma unroll
    for (int rep = 0; rep < 8; rep++) {
      int idx = rep * 256 + t;             // float4 index over 128 x 16
      int row = idx >> 4, c4 = idx & 15;
      *(float4*)&SW[swbase + (size_t)(i0 + row) * Tt + j0 + c4 * 4] =
          make_float4(0.f, 0.f, 0.f, 0.f);
    }
    return;
  }

  const float* qb = q + (size_t)b * Tt * Kk;
  const float* kb = kv + (size_t)b * Tt * Kk;
  int w = t >> 5, l = t & 31, half = l >> 4, lr = l & 15;
  v8f acc[4] = {};

  for (int k0 = 0; k0 < Kk; k0 += BKc) {
    __syncthreads();
    #pragma unroll
    for (int rep = 0; rep < 4; rep++) {    // A: 128 rows x 8 float4
      int idx = rep * 256 + t;
      int row = idx >> 3, c4 = idx & 7;
      async_b128(lds_off(&As[row * LDA + c4 * 4]),
                 (uint64_t)(uintptr_t)(qb + (size_t)(i0 + row) * Kk + k0 + c4 * 4));
    }
    #pragma unroll
    for (int rep = 0; rep < 2; rep++) {    // B: 64 rows x 8 float4
      int idx = rep * 256 + t;
      int row = idx >> 3, c4 = idx & 7;
      async_b128(lds_off(&Bs[row * LDA + c4 * 4]),
                 (uint64_t)(uintptr_t)(kb + (size_t)(j0 + row) * Kk + k0 + c4 * 4));
    }
    wait_async();
    __syncthreads();
    #pragma unroll
    for (int k = 0; k < BKc; k += 4) {
      int ka = k + 2 * half;
      v2f a;
      a.x = As[(w * 16 + lr) * LDA + ka];
      a.y = As[(w * 16 + lr) * LDA + ka + 1];
      #pragma unroll
      for (int c = 0; c < 4; c++) {        // B fragment: B[k][n] = kk[j0+n][k]
        v2f bf;
        bf.x = Bs[(c * 16 + lr) * LDA + ka];
        bf.y = Bs[(c * 16 + lr) * LDA + ka + 1];
        acc[c] = WMMA_F32(a, bf, acc[c]);
      }
    }
  }

  float decay = 1.f / (1.f + expf(-dlogit[0]));
  float l2d = log2f(decay);
  #pragma unroll
  for (int c = 0; c < 4; c++) {
    #pragma unroll
    for (int r = 0; r < 8; r++) {
      int i = i0 + w * 16 + r + 8 * half;
      int j = j0 + c * 16 + lr;
      float wt = (j > i) ? exp2f(l2d * (float)(j - i - 1)) : 0.f;
      SW[swbase + (size_t)i * Tt + j] = acc[c][r] * wt * 0.0625f;   // 1/sqrt(256)
    }
  }
}

// ---------------- retrieved = SW @ V, LDS-tiled, K-loop starts at the diagonal ----------------
__global__ void __launch_bounds__(256) k_retr(const float* __restrict__ SW,
                                              const float* __restrict__ vv,
                                              float* __restrict__ retr) {
  __shared__ float As[BI * LDA];           // 128 x 36  (SW rows, k-contiguous)
  __shared__ float Bs[BKc * LDBR];         // 32 x 68   (vv rows, n-contiguous)
  const int ITG = Tt / BI;                 // 16
  const int NTG = Kk / RBN;                // 4
  int id = blockIdx.x;
  int b = id / (ITG * NTG);
  int rem = id % (ITG * NTG);
  int itg = rem / NTG, ntg = rem % NTG;
  int i0 = itg * BI, n0 = ntg * RBN;
  int t = threadIdx.x;
  const float* sw = SW + (size_t)b * Tt * Tt;
  const float* vb = vv + (size_t)b * Tt * Kk;
  int w = t >> 5, l = t & 31, half = l >> 4, lr = l & 15;
  v8f acc[4] = {};

  for (int k0 = i0; k0 < Tt; k0 += BKc) {  // SW is zero left of the diagonal block-row
    __syncthreads();
    #pragma unroll
    for (int rep = 0; rep < 4; rep++) {    // A: 128 rows x 8 float4
      int idx = rep * 256 + t;
      int row = idx >> 3, c4 = idx & 7;
      async_b128(lds_off(&As[row * LDA + c4 * 4]),
                 (uint64_t)(uintptr_t)(sw + (size_t)(i0 + row) * Tt + k0 + c4 * 4));
    }
    #pragma unroll
    for (int rep = 0; rep < 2; rep++) {    // B: 32 rows (k) x 16 float4 (n)
      int idx = rep * 256 + t;
      int row = idx >> 4, c4 = idx & 15;
      async_b128(lds_off(&Bs[row * LDBR + c4 * 4]),
                 (uint64_t)(uintptr_t)(vb + (size_t)(k0 + row) * Kk + n0 + c4 * 4));
    }
    wait_async();
    __syncthreads();
    #pragma unroll
    for (int k = 0; k < BKc; k += 4) {
      int ka = k + 2 * half;
      v2f a;
      a.x = As[(w * 16 + lr) * LDA + ka];
      a.y = As[(w * 16 + lr) * LDA + ka + 1];
      #pragma unroll
      for (int c = 0; c < 4; c++) {        // B fragment: B[k][n] = vv[k][n]
        v2f bf;
        bf.x = Bs[ka * LDBR + c * 16 + lr];
        bf.y = Bs[(ka + 1) * LDBR + c * 16 + lr];
        acc[c] = WMMA_F32(a, bf, acc[c]);
      }
    }
  }

  float* rb = retr + (size_t)b * Tt * Kk;
  #pragma unroll
  for (int c = 0; c < 4; c++) {
    #pragma unroll
    for (int r = 0; r < 8; r++) {
      int mm = i0 + w * 16 + r + 8 * half;
      rb[(size_t)mm * Kk + n0 + c * 16 + lr] = acc[c][r];
    }
  }
}

// ---------------- g2 = g + mem_scale*out_scale*(retr @ Wo) ----------------
__global__ void k_memres(const float* __restrict__ retr, const float* __restrict__ Wo,
                         const float* __restrict__ g, const float* __restrict__ oscale,
                         const float* __restrict__ mscale, float* __restrict__ g2) {
  const int Nt = Kk / 16;
  int id = blockIdx.x;
  int mt = id / Nt, nt = id % Nt;
  int l = threadIdx.x, half = l >> 4, lr = l & 15;
  int m = mt * 16 + lr, n = nt * 16 + lr;
  v8f acc = {};
  #pragma unroll 4
  for (int k0 = 0; k0 < Kk; k0 += 4) {
    int ka = k0 + 2 * half;
    v2f a;  a.x = retr[(size_t)m * Kk + ka]; a.y = retr[(size_t)m * Kk + ka + 1];
    v2f bf; bf.x = Wo[(size_t)ka * Kk + n];  bf.y = Wo[(size_t)(ka + 1) * Kk + n];
    acc = WMMA_F32(a, bf, acc);
  }
  float coef = oscale[0] * mscale[0];
  #pragma unroll
  for (int r = 0; r < 8; r++) {
    int mm = mt * 16 + r + 8 * half;
    g2[(size_t)mm * Kk + n] = g[(size_t)mm * Kk + n] + coef * acc[r];
  }
}

// ---------------- rmsnorm ----------------
__global__ void k_norm(const float* __restrict__ g2, float* __restrict__ gn2) {
  int row = blockIdx.x, k = threadIdx.x;
  float val = g2[(size_t)row * Kk + k];
  __shared__ float red[256];
  red[k] = val * val;
  __syncthreads();
  for (int s = 128; s > 0; s >>= 1) { if (k < s) red[k] += red[k + s]; __syncthreads(); }
  gn2[(size_t)row * Kk + k] = val * rsqrtf(red[0] / (float)Kk + EPSf);
}

// ---------------- H = gelu(gn2 @ branch_down + bias), branches concatenated (N=512) ----------------
__global__ void k_down(const float* __restrict__ gn2, const float* __restrict__ BD,
                       const float* __restrict__ bias, float* __restrict__ H) {
  const int Nt = (NBr * INNERd) / 16;      // 32
  int id = blockIdx.x;
  int mt = id / Nt, nt = id % Nt;
  int l = threadIdx.x, half = l >> 4, lr = l & 15;
  int m = mt * 16 + lr;
  int c = nt * 16 + lr;                    // 0..511
  int nb = c >> 7, h = c & 127;
  v8f acc = {};
  #pragma unroll 4
  for (int k0 = 0; k0 < Kk; k0 += 4) {
    int ka = k0 + 2 * half;
    v2f a;  a.x = gn2[(size_t)m * Kk + ka]; a.y = gn2[(size_t)m * Kk + ka + 1];
    v2f bf;
    bf.x = BD[((size_t)nb * Kk + ka) * INNERd + h];
    bf.y = BD[((size_t)nb * Kk + ka + 1) * INNERd + h];
    acc = WMMA_F32(a, bf, acc);
  }
  float bi = bias[h];
  #pragma unroll
  for (int r = 0; r < 8; r++) {
    int mm = mt * 16 + r + 8 * half;
    float xv = acc[r] + bi;
    H[(size_t)mm * (NBr * INNERd) + c] = 0.5f * xv * (1.0f + erff(xv * 0.70710678118654752f));
  }
}

// ---------------- bout[n] = H[:,n] @ branch_up[n] ----------------
__global__ void k_up(const float* __restrict__ H, const float* __restrict__ BU,
                     float* __restrict__ bout) {
  const int Mt = Mrows / 16, Nt = Kk / 16;
  int id = blockIdx.x;
  int nb = id / (Mt * Nt);
  int rem = id % (Mt * Nt);
  int mt = rem / Nt, nt = rem % Nt;
  int l = threadIdx.x, half = l >> 4, lr = l & 15;
  int m = mt * 16 + lr, n = nt * 16 + lr;
  const int ldh = NBr * INNERd;
  v8f acc = {};
  #pragma unroll 4
  for (int k0 = 0; k0 < INNERd; k0 += 4) {
    int ka = k0 + 2 * half;
    v2f a;  a.x = H[(size_t)m * ldh + nb * INNERd + ka];
            a.y = H[(size_t)m * ldh + nb * INNERd + ka + 1];
    v2f bf; bf.x = BU[((size_t)nb * INNERd + ka) * Kk + n];
            bf.y = BU[((size_t)nb * INNERd + ka + 1) * Kk + n];
    acc = WMMA_F32(a, bf, acc);
  }
  #pragma unroll
  for (int r = 0; r < 8; r++) {
    int mm = mt * 16 + r + 8 * half;
    bout[((size_t)mm * NBr + nb) * Kk + n] = acc[r];
  }
}

// ---------------- gates = softmax(gn2@gate_W + gate_b); outk = sum_n gates*bout ----------------
__global__ void k_combine(const float* __restrict__ gn2, const float* __restrict__ bout,
                          const float* __restrict__ gW, const float* __restrict__ gB,
                          const float* __restrict__ wscale, float* __restrict__ outk) {
  int row = blockIdx.x, t = threadIdx.x;
  __shared__ float red[NBr][256];
  float gv = gn2[(size_t)row * Kk + t];
  #pragma unroll
  for (int n = 0; n < NBr; n++) red[n][t] = gv * gW[(size_t)t * NBr + n];
  __syncthreads();
  for (int s = 128; s > 0; s >>= 1) {
    if (t < s) {
      #pragma unroll
      for (int n = 0; n < NBr; n++) red[n][t] += red[n][t + s];
    }
    __syncthreads();
  }
  float logits[NBr], ex[NBr], mx = -1e30f, se = 0.f;
  #pragma unroll
  for (int n = 0; n < NBr; n++) { logits[n] = red[n][0] + gB[n]; mx = fmaxf(mx, logits[n]); }
  #pragma unroll
  for (int n = 0; n < NBr; n++) { ex[n] = expf(logits[n] - mx); se += ex[n]; }
  float scale = wscale[0] * 0.0625f;       // write_scale / sqrt(256)
  float acc = 0.f;
  #pragma unroll
  for (int n = 0; n < NBr; n++) acc += (ex[n] / se) * bout[((size_t)row * NBr + n) * Kk + t];
  outk[(size_t)row * Kk + t] = acc * scale;
}

// ---------------- output: zero-fill + scatter ----------------
__global__ void k_zero(float4* __restrict__ out, size_t n4) {
  size_t i = (size_t)blockIdx.x * blockDim.x + threadIdx.x;
  if (i < n4) out[i] = make_float4(0.f, 0.f, 0.f, 0.f);
}

__global__ void k_scatter(const float* __restrict__ outk, const int* __restrict__ widx,
                          float* __restrict__ out) {
  int row = blockIdx.x, k = threadIdx.x;
  out[(size_t)row * Vv + widx[k]] = outk[(size_t)row * Kk + k];
}

extern "C" void kernel_launch(void* const* d_in, const int* in_sizes, int n_in,
                              void* d_out, int out_size, void* d_ws, size_t ws_size,
                              hipStream_t stream) {
  (void)in_sizes; (void)n_in; (void)ws_size;
  const float* x      = (const float*)d_in[0];
  const float* Wq     = (const float*)d_in[1];
  const float* Wk     = (const float*)d_in[2];
  const float* Wv     = (const float*)d_in[3];
  const float* Wo     = (const float*)d_in[4];
  const float* dlogit = (const float*)d_in[5];
  const float* oscale = (const float*)d_in[6];
  const float* mscale = (const float*)d_in[7];
  const float* BD     = (const float*)d_in[8];
  const float* BU     = (const float*)d_in[9];
  const float* bias   = (const float*)d_in[10];
  const float* gW     = (const float*)d_in[11];
  const float* gB     = (const float*)d_in[12];
  const float* wscale = (const float*)d_in[13];
  const int*   ridx   = (const int*)d_in[14];
  const int*   widx   = (const int*)d_in[15];
  float* out = (float*)d_out;

  float* ws = (float*)d_ws;
  const size_t U = (size_t)Mrows * Kk;     // 1,048,576 floats (4 MB)
  float* g    = ws + 0 * U;
  float* gn   = ws + 1 * U;
  float* q    = ws + 2 * U;
  float* kk   = ws + 3 * U;
  float* vv   = ws + 4 * U;
  float* g2   = ws + 5 * U;
  float* gn2  = ws + 6 * U;
  float* outk = ws + 7 * U;
  float* retr = ws + 8 * U;
  float* H    = ws + 9 * U;                // 2U  (B*T x 512)
  float* bout = ws + 11 * U;               // 4U  (B*T x NB x K)
  float* SW   = ws + 15 * U;               // 8U  (B x T x T) -> 23U total = 92 MB

  k_gather_norm<<<Mrows, 256, 0, stream>>>(x, ridx, g, gn);
  k_qkv<<<3 * (Mrows / 16) * (Kk / 16), 32, 0, stream>>>(gn, Wq, Wk, Wv, q, kk, vv);
  k_scores<<<Bb * (Tt / BI) * (Tt / BJ), 256, 0, stream>>>(q, kk, dlogit, SW);
  k_retr<<<Bb * (Tt / BI) * (Kk / RBN), 256, 0, stream>>>(SW, vv, retr);
  k_memres<<<(Mrows / 16) * (Kk / 16), 32, 0, stream>>>(retr, Wo, g, oscale, mscale, g2);
  k_norm<<<Mrows, 256, 0, stream>>>(g2, gn2);
  k_down<<<(Mrows / 16) * ((NBr * INNERd) / 16), 32, 0, stream>>>(gn2, BD, bias, H);
  k_up<<<NBr * (Mrows / 16) * (Kk / 16), 32, 0, stream>>>(H, BU, bout);
  k_combine<<<Mrows, 256, 0, stream>>>(gn2, bout, gW, gB, wscale, outk);
  size_t n4 = (size_t)out_size / 4;
  k_zero<<<(unsigned)((n4 + 255) / 256), 256, 0, stream>>>((float4*)out, n4);
  k_scatter<<<Mrows, 256, 0, stream>>>(outk, widx, out);
}